// SPDBatchNormImpl_50577534878042
// MI455X (gfx1250) — compile-verified
//
#include <hip/hip_runtime.h>
#include <hip/hip_bf16.h>

// ---------------------------------------------------------------------------
// SPD batch-norm (training fwd) for 32x32 SPD matrices on gfx1250 (MI455X).
// All matrix functions are matmul-only iterations running on
// V_WMMA_F32_16X16X4_F32; one wave32 owns one 32x32 matrix in LDS.
// Every B operand in the pipeline is symmetric (polynomials of one symmetric
// matrix), so B fragments are read as contiguous row pairs (ds_load_b64),
// and all affine epilogues (alpha*(A*B) + beta*I) are folded into the WMMA
// accumulator at store time. All buffer routing is compile-time static.
// ---------------------------------------------------------------------------

typedef __attribute__((ext_vector_type(2))) float v2f;
typedef __attribute__((ext_vector_type(8))) float v8f;

#define MATDIM 32
#define NP     34               // padded LDS row stride: even -> 8B-aligned pairs
#define MATP   (MATDIM * NP)
#define MATN   (MATDIM * MATDIM)
#define NBUF   5                // LDS matrices per wave
#define WPB    2                // waves per block
#define TPB    (WPB * 32)

#define NS_ITERS 12             // Newton-Schulz iterations for sqrt/invsqrt
#define NSQRT    4              // inverse-scaling-squaring roots in logm
#define NLOG     8              // Mercator series terms
#define NEXP     8              // exp Taylor terms
#define MEAN_SPLIT 32           // partial-sum groups for batch reductions
#define BN_EPS_C 1e-5f

// ns_sqrt's internal (y,z,u) pointer rotation has period 3. After
// (NS_ITERS-1) loop iterations, sqrt/invsqrt land in statically known args:
//   r = (NS_ITERS-1) % 3 :  r==0 -> sqrt in y-arg, invsqrt in z-arg
//                           r==1 -> sqrt in u-arg, invsqrt in y-arg
//                           r==2 -> sqrt in z-arg, invsqrt in u-arg
#define NS_R    ((NS_ITERS - 1) % 3)
#define SQ_IDX  (NS_R == 0 ? 0 : NS_R == 1 ? 3 : 1)   // index into {y,z,t,u}
#define ISQ_IDX (NS_R == 0 ? 1 : NS_R == 1 ? 0 : 3)
static_assert(NS_ITERS >= 3, "need at least 3 NS iterations (peeled last)");

// Wave-local fence: this wave's LDS stores visible before the next LDS read
// (cross-lane producer/consumer inside one wave) + compiler reorder barrier.
__device__ __forceinline__ void lds_fence() {
    asm volatile("s_wait_dscnt 0" ::: "memory");
}

__device__ __forceinline__ float wave_sum(float v) {
    for (int o = 16; o > 0; o >>= 1) v += __shfl_xor(v, o, 32);
    return v;
}

// ---------------------------------------------------------------------------
// D = alpha*(A*B) + beta*I for 32x32 fp32, one wave; B must be symmetric.
// 2x2 tiles of 16x16, K in steps of 4 via V_WMMA_F32_16X16X4_F32.
// A 16x4 frag (ISA 7.12.2): lanes 0-15 M=lane K{0,1}; lanes 16-31 K{2,3}.
// With B symmetric, B[k][n]=B[n][k]: both A and B fragments are contiguous
// row pairs -> ds_load_b64, no cross-register repacking.
// C/D: VGPR r -> M = r + 8*(lane/16), N = lane%16.
// ---------------------------------------------------------------------------
__device__ __forceinline__ void mm32s(const float* A, const float* Bs, float* D,
                                      float alpha, float beta, int lane) {
    const int h = lane >> 4;
    const int l = lane & 15;
    v8f acc00 = {}, acc01 = {}, acc10 = {}, acc11 = {};
#pragma unroll
    for (int k = 0; k < 8; ++k) {
        const int kk = 4 * k + 2 * h;                   // even -> 8B aligned
        v2f a0 = *(const v2f*)(A  + l        * NP + kk);
        v2f a1 = *(const v2f*)(A  + (l + 16) * NP + kk);
        v2f b0 = *(const v2f*)(Bs + l        * NP + kk); // = B[kk..kk+1][l]
        v2f b1 = *(const v2f*)(Bs + (l + 16) * NP + kk); // = B[kk..kk+1][l+16]
        acc00 = __builtin_amdgcn_wmma_f32_16x16x4_f32(false, a0, false, b0,
                                                      (short)0, acc00, false, false);
        acc01 = __builtin_amdgcn_wmma_f32_16x16x4_f32(false, a0, false, b1,
                                                      (short)0, acc01, false, false);
        acc10 = __builtin_amdgcn_wmma_f32_16x16x4_f32(false, a1, false, b0,
                                                      (short)0, acc10, false, false);
        acc11 = __builtin_amdgcn_wmma_f32_16x16x4_f32(false, a1, false, b1,
                                                      (short)0, acc11, false, false);
    }
#pragma unroll
    for (int r = 0; r < 8; ++r) {
        const int m = r + 8 * h;
        const float diag = (m == l) ? beta : 0.f;       // only 00/11 tiles
        D[m * NP + l]              = alpha * acc00[r] + diag;
        D[m * NP + l + 16]         = alpha * acc01[r];
        D[(m + 16) * NP + l]       = alpha * acc10[r];
        D[(m + 16) * NP + l + 16]  = alpha * acc11[r] + diag;
    }
    lds_fence();
}

// ------------------------- element-wise LDS helpers ------------------------
__device__ __forceinline__ void ld_mat(float* dst, const float* __restrict__ src,
                                       int lane) {
    const float4* s = (const float4*)(src + (size_t)lane * MATDIM);
    float* drow = dst + lane * NP;
#pragma unroll
    for (int t = 0; t < 8; ++t) {
        float4 v = s[t];
        v2f p0; p0.x = v.x; p0.y = v.y;
        v2f p1; p1.x = v.z; p1.y = v.w;
        *(v2f*)(drow + 4 * t)     = p0;
        *(v2f*)(drow + 4 * t + 2) = p1;
    }
    lds_fence();
}
__device__ __forceinline__ void st_mat(float* __restrict__ dst, const float* src,
                                       int lane) {
    float4* d = (float4*)(dst + (size_t)lane * MATDIM);
    const float* srow = src + lane * NP;
#pragma unroll
    for (int t = 0; t < 8; ++t) {
        v2f p0 = *(const v2f*)(srow + 4 * t);
        v2f p1 = *(const v2f*)(srow + 4 * t + 2);
        float4 v; v.x = p0.x; v.y = p0.y; v.z = p1.x; v.w = p1.y;
        d[t] = v;
    }
}
__device__ __forceinline__ void set_ident(float* dst, float d, int lane) {
    float* row = dst + lane * NP;
#pragma unroll
    for (int t = 0; t < 16; ++t) {
        v2f p;
        p.x = (2 * t     == lane) ? d : 0.f;
        p.y = (2 * t + 1 == lane) ? d : 0.f;
        *(v2f*)(row + 2 * t) = p;
    }
    lds_fence();
}
// dst = a*src + b*I  (row per lane; dst may alias src)
__device__ __forceinline__ void axpbI(float* dst, const float* src, float a,
                                      float b, int lane) {
    const float* srow = src + lane * NP;
    float* drow = dst + lane * NP;
#pragma unroll
    for (int t = 0; t < 16; ++t) {
        v2f v = *(const v2f*)(srow + 2 * t);
        v.x = a * v.x + ((2 * t     == lane) ? b : 0.f);
        v.y = a * v.y + ((2 * t + 1 == lane) ? b : 0.f);
        *(v2f*)(drow + 2 * t) = v;
    }
    lds_fence();
}
__device__ __forceinline__ float fro_norm(const float* src, int lane) {
    const float* srow = src + lane * NP;
    float s = 0.f;
#pragma unroll
    for (int t = 0; t < 16; ++t) {
        v2f v = *(const v2f*)(srow + 2 * t);
        s += v.x * v.x + v.y * v.y;
    }
    return sqrtf(wave_sum(s));
}

// ---------------------------------------------------------------------------
// Coupled Newton-Schulz: sqrt(M)+ybeta*I -> arg SQ_IDX, invsqrt(M) -> ISQ_IDX.
// M read-only; first iteration specialized (Z0=I); last iteration peeled with
// the sqrt(s)/1/sqrt(s) rescales (and ybeta) folded into its alphas.
// ---------------------------------------------------------------------------
__device__ __forceinline__ void ns_sqrt(const float* M, float* y, float* z,
                                        float* t, float* u, float ybeta,
                                        int lane) {
    const float s  = fmaxf(fro_norm(M, lane), 1e-12f);
    const float rs = sqrtf(s);
    axpbI(z, M, -0.5f / s, 1.5f, lane);                 // z = G1
    mm32s(M, z, y, 1.0f / s, 0.f, lane);                // y = Y1 = (M/s)*G1
    for (int it = 1; it < NS_ITERS - 1; ++it) {
        mm32s(z, y, t, -0.5f, 1.5f, lane);              // G = 1.5I - 0.5*Z*Y
        mm32s(y, t, u, 1.f, 0.f, lane);                 // Y' = Y*G
        { float* p = y; y = u; u = p; }
        mm32s(t, z, u, 1.f, 0.f, lane);                 // Z' = G*Z
        { float* p = z; z = u; u = p; }
    }
    // peeled last iteration with folded output scaling
    mm32s(z, y, t, -0.5f, 1.5f, lane);
    mm32s(y, t, u, rs, ybeta, lane);                    // sqrt(M) + ybeta*I
    { float* p = y; y = u; u = p; }
    mm32s(t, z, u, 1.f / rs, 0.f, lane);                // invsqrt(M)
    { float* p = z; z = u; u = p; }
}

// ---------------------------------------------------------------------------
// logm of SPD matrix in m; scratch w1..w4, all routing compile-time static.
// NSQRT repeated NS roots (last emits E = sqrt - I directly), then
// log(I+E) = E * p(E) via fully unrolled Horner with constant coefficients,
// final 2^NSQRT scale folded into the last matmul. Returns result pointer
// (compile-time constant after inlining).
// ---------------------------------------------------------------------------
__device__ __forceinline__ float* logm32(float* m, float* w1, float* w2,
                                         float* w3, float* w4, int lane) {
    float* s[4] = {w1, w2, w3, w4};
#pragma unroll
    for (int q = 0; q < NSQRT; ++q) {
        const float yb = (q == NSQRT - 1) ? -1.f : 0.f;
        ns_sqrt(m, s[0], s[1], s[2], s[3], yb, lane);
        float* sq = s[SQ_IDX];                           // constant index
        s[SQ_IDX] = m;
        m = sq;
    }
    // m now holds E = M^(1/2^NSQRT) - I ; p(E) coefficients (-1)^i/(i+1)
    const float CLOG[NLOG] = {1.f, -1.f / 2, 1.f / 3, -1.f / 4,
                              1.f / 5, -1.f / 6, 1.f / 7, -1.f / 8};
    set_ident(s[0], CLOG[NLOG - 1], lane);
    float* P = s[0]; float* T = s[1];
#pragma unroll
    for (int i = NLOG - 2; i >= 0; --i) {
        mm32s(m, P, T, 1.f, CLOG[i], lane);             // P' = ci*I + E*P
        { float* p = P; P = T; T = p; }
    }
    mm32s(m, P, T, (float)(1 << NSQRT), 0.f, lane);     // L = 2^k * E*p(E)
    return T;
}

// ---------------------------------------------------------------------------
// expm(pre * a): a read-only, scratch w1,w2. Scaling & squaring; `pre` and
// 2^-j folded into the (unrolled) Horner alphas; j from frexpf (no loop).
// ---------------------------------------------------------------------------
__device__ __forceinline__ float* expm32(const float* a, float* w1, float* w2,
                                         float pre, int lane) {
    const float f = fro_norm(a, lane) * fabsf(pre);
    int e;
    (void)frexpf(f, &e);                 // f = m*2^e, m in [0.5,1)
    const int j = (f > 0.25f) ? max(0, e + 2) : 0;      // f*2^-j <= 0.25
    const float inv = ldexpf(pre, -j);
    const float IF[NEXP + 1] = {1.f, 1.f, 1.f / 2, 1.f / 6, 1.f / 24,
                                1.f / 120, 1.f / 720, 1.f / 5040, 1.f / 40320};
    set_ident(w1, IF[NEXP], lane);
    float* P = w1; float* T = w2;
#pragma unroll
    for (int i = NEXP - 1; i >= 0; --i) {
        mm32s(a, P, T, inv, IF[i], lane);               // P' = I/i! + Es*P
        { float* p = P; P = T; T = p; }
    }
    for (int q = 0; q < j; ++q) {
        mm32s(P, P, T, 1.f, 0.f, lane);
        { float* p = P; P = T; T = p; }
    }
    return P;
}

// =============================== kernels ===================================

__global__ void k_mean_part(const float* __restrict__ X, float* __restrict__ part,
                            int Bn, int Cn) {
    long idx = (long)blockIdx.x * blockDim.x + threadIdx.x;
    long tot = (long)Cn * MATN * MEAN_SPLIT;
    if (idx >= tot) return;
    int g = (int)(idx / ((long)Cn * MATN));
    int rem = (int)(idx % ((long)Cn * MATN));
    int per = (Bn + MEAN_SPLIT - 1) / MEAN_SPLIT;
    int b0 = g * per, b1 = min(Bn, b0 + per);
    float s = 0.f;
    for (int b = b0; b < b1; ++b)
        s += X[(size_t)b * Cn * MATN + rem];
    part[idx] = s;
}

__global__ void k_mean_fin(const float* __restrict__ part, float* __restrict__ out,
                           int Bn, int Cn) {
    int idx = blockIdx.x * blockDim.x + threadIdx.x;
    if (idx >= Cn * MATN) return;
    float s = 0.f;
    for (int g = 0; g < MEAN_SPLIT; ++g)
        s += part[(long)g * Cn * MATN + idx];
    out[idx] = s / (float)Bn;
}

// Per-channel sqrt/invsqrt (Newton-Schulz), one wave per channel.
__global__ void k_sqrt_isqrt(const float* __restrict__ in,
                             float* __restrict__ sq_out,
                             float* __restrict__ isq_out, int Cn) {
    __shared__ float lds[WPB][NBUF][MATP];
    const int wave = threadIdx.x >> 5, lane = threadIdx.x & 31;
    const int c = blockIdx.x * WPB + wave;
    if (c >= Cn) return;
    float* b0 = lds[wave][0];
    float* s[4] = {lds[wave][1], lds[wave][2], lds[wave][3], lds[wave][4]};
    ld_mat(b0, in + (size_t)c * MATN, lane);
    ns_sqrt(b0, s[0], s[1], s[2], s[3], 0.f, lane);
    st_mat(sq_out + (size_t)c * MATN, s[SQ_IDX], lane);
    st_mat(isq_out + (size_t)c * MATN, s[ISQ_IDX], lane);
}

// XT = logm(bm_isq * X * bm_isq), one wave per (b,c) sample.
__global__ void k_log_tangent(const float* __restrict__ X,
                              const float* __restrict__ isq_g,
                              float* __restrict__ XT, long nmat, int Cn) {
    __shared__ float lds[WPB][NBUF][MATP];
    const int wave = threadIdx.x >> 5, lane = threadIdx.x & 31;
    const long gid = (long)blockIdx.x * WPB + wave;
    if (gid >= nmat) return;
    const int c = (int)(gid % Cn);
    float* b0 = lds[wave][0]; float* b1 = lds[wave][1];
    float* b2 = lds[wave][2]; float* b3 = lds[wave][3];
    float* b4 = lds[wave][4];
    ld_mat(b0, X + gid * MATN, lane);
    ld_mat(b1, isq_g + (size_t)c * MATN, lane);
    mm32s(b1, b0, b2, 1.f, 0.f, lane);       // isq * X        (B = X sym)
    mm32s(b2, b1, b3, 1.f, 0.f, lane);       // (isq X) * isq  (B = isq sym)
    float* L = logm32(b3, b0, b1, b2, b4, lane);
    st_mat(XT + gid * MATN, L, lane);
}

// Per-channel: rm = bm_sq * expm(GT) * bm_sq ; store invsqrt(rm).
__global__ void k_rm_isq(const float* __restrict__ GT,
                         const float* __restrict__ bm_sq,
                         float* __restrict__ rm_isq_out, int Cn) {
    __shared__ float lds[WPB][NBUF][MATP];
    const int wave = threadIdx.x >> 5, lane = threadIdx.x & 31;
    const int c = blockIdx.x * WPB + wave;
    if (c >= Cn) return;
    float* b0 = lds[wave][0]; float* b1 = lds[wave][1];
    float* b2 = lds[wave][2]; float* b3 = lds[wave][3];
    float* b4 = lds[wave][4];
    ld_mat(b0, GT + (size_t)c * MATN, lane);
    float* E = expm32(b0, b1, b2, 1.f, lane);        // E in b1 or b2
    ld_mat(b3, bm_sq + (size_t)c * MATN, lane);
    mm32s(b3, E, b4, 1.f, 0.f, lane);                // bm_sq * E   (B = E sym)
    mm32s(b4, b3, b0, 1.f, 0.f, lane);               // rm          (B sym)
    float* s[4] = {b1, b2, b3, b4};
    ns_sqrt(b0, s[0], s[1], s[2], s[3], 0.f, lane);
    st_mat(rm_isq_out + (size_t)c * MATN, s[ISQ_IDX], lane);
}

// batch_var[c] = mean_B ||XT - GT||_F^2 (GTrm == GT exactly since ETA=1).
__global__ void k_var(const float* __restrict__ XT, const float* __restrict__ GT,
                      float* __restrict__ var, long nmat, int Bn, int Cn) {
    const int wave = threadIdx.x >> 5, lane = threadIdx.x & 31;
    const long gid = (long)blockIdx.x * (blockDim.x >> 5) + wave;
    if (gid >= nmat) return;
    const int c = (int)(gid % Cn);
    const float* xt = XT + gid * MATN;
    const float* gt = GT + (size_t)c * MATN;
    float s = 0.f;
    for (int i = lane; i < MATN; i += 32) {
        float d = xt[i] - gt[i];
        s += d * d;
    }
    s = wave_sum(s);
    if (lane == 0) atomicAdd(&var[c], s / (float)Bn);
}

// s[c] = std[c] / sqrt(batch_var[c] + BN_EPS)   (ETA=1 -> rv = batch_var)
__global__ void k_scale(const float* __restrict__ var, const float* __restrict__ stdv,
                        float* __restrict__ s_out, int Cn) {
    int c = threadIdx.x;
    if (c < Cn) s_out[c] = stdv[c] * rsqrtf(var[c] + BN_EPS_C);
}

// Xn = mean_sq * expm(s * logm(rm_isq * X * rm_isq)) * mean_sq
__global__ void k_final(const float* __restrict__ X,
                        const float* __restrict__ rm_isq,
                        const float* __restrict__ mean_sq,
                        const float* __restrict__ s_c,
                        float* __restrict__ out, long nmat, int Cn) {
    __shared__ float lds[WPB][NBUF][MATP];
    const int wave = threadIdx.x >> 5, lane = threadIdx.x & 31;
    const long gid = (long)blockIdx.x * WPB + wave;
    if (gid >= nmat) return;
    const int c = (int)(gid % Cn);
    float* b0 = lds[wave][0]; float* b1 = lds[wave][1];
    float* b2 = lds[wave][2]; float* b3 = lds[wave][3];
    float* b4 = lds[wave][4];
    ld_mat(b0, X + gid * MATN, lane);
    ld_mat(b1, rm_isq + (size_t)c * MATN, lane);
    mm32s(b1, b0, b2, 1.f, 0.f, lane);               // isq * X
    mm32s(b2, b1, b3, 1.f, 0.f, lane);               // inner = isq*X*isq
    float* L = logm32(b3, b0, b1, b2, b4, lane);     // constant ptr after inline
    // compile-time selection of the buffers not holding L
    float* all[5] = {b0, b1, b2, b3, b4};
    float* fr[4]; int t = 0;
#pragma unroll
    for (int i = 0; i < 5; ++i)
        if (all[i] != L) fr[t++] = all[i];
    float* P = expm32(L, fr[0], fr[1], s_c[c], lane);     // (inner)^s
    ld_mat(fr[2], mean_sq + (size_t)c * MATN, lane);
    mm32s(fr[2], P, fr[3], 1.f, 0.f, lane);          // mean_sq * P (B = P sym)
    mm32s(fr[3], fr[2], L, 1.f, 0.f, lane);          // * mean_sq   (B sym)
    st_mat(out + gid * MATN, L, lane);
}

// ================================ host =====================================

extern "C" void kernel_launch(void* const* d_in, const int* in_sizes, int n_in,
                              void* d_out, int out_size, void* d_ws, size_t ws_size,
                              hipStream_t stream) {
    const float* X     = (const float*)d_in[0];  // [B,C,32,32]
    const float* rmean = (const float*)d_in[1];  // unused: ETA=1 -> rm=batch_mean
    const float* rvar  = (const float*)d_in[2];  // unused: ETA=1 -> rv=batch_var
    const float* meanP = (const float*)d_in[3];  // [C,32,32]
    const float* stdP  = (const float*)d_in[4];  // [C]
    (void)rmean; (void)rvar; (void)n_in; (void)out_size; (void)ws_size;

    const int  Cn = in_sizes[4];
    const int  Bn = in_sizes[0] / (Cn * MATN);
    const long nmat = (long)Bn * Cn;

    float* ws = (float*)d_ws;
    size_t off = 0;
    float* bm       = ws + off; off += (size_t)Cn * MATN;
    float* bm_sq    = ws + off; off += (size_t)Cn * MATN;
    float* bm_isq   = ws + off; off += (size_t)Cn * MATN;
    float* GT       = ws + off; off += (size_t)Cn * MATN;
    float* rm_isq   = ws + off; off += (size_t)Cn * MATN;
    float* mean_sq  = ws + off; off += (size_t)Cn * MATN;
    float* mean_isq = ws + off; off += (size_t)Cn * MATN;   // dummy output
    float* var      = ws + off; off += (size_t)Cn;
    float* svec     = ws + off; off += (size_t)Cn;
    float* part     = ws + off; off += (size_t)MEAN_SPLIT * Cn * MATN;
    float* XT       = ws + off; off += (size_t)nmat * MATN;

    const int redT = 256;
    const long partN = (long)Cn * MATN * MEAN_SPLIT;
    const unsigned partB = (unsigned)((partN + redT - 1) / redT);
    const unsigned finB  = (unsigned)((Cn * MATN + redT - 1) / redT);
    const unsigned chB   = (unsigned)((Cn + WPB - 1) / WPB);
    const unsigned smB   = (unsigned)((nmat + WPB - 1) / WPB);

    // 1) batch_mean = mean_B X
    k_mean_part<<<partB, redT, 0, stream>>>(X, part, Bn, Cn);
    k_mean_fin<<<finB, redT, 0, stream>>>(part, bm, Bn, Cn);
    // 2) bm_sq / bm_isq per channel
    k_sqrt_isqrt<<<chB, TPB, 0, stream>>>(bm, bm_sq, bm_isq, Cn);
    // 3) XT = logm(bm_isq X bm_isq) per sample
    k_log_tangent<<<smB, TPB, 0, stream>>>(X, bm_isq, XT, nmat, Cn);
    // 4) GT = mean_B XT
    k_mean_part<<<partB, redT, 0, stream>>>(XT, part, Bn, Cn);
    k_mean_fin<<<finB, redT, 0, stream>>>(part, GT, Bn, Cn);
    // 5) rm = bm_sq expm(GT) bm_sq ; rm_isq = invsqrt(rm) ; mean_sq = sqrt(mean)
    k_rm_isq<<<chB, TPB, 0, stream>>>(GT, bm_sq, rm_isq, Cn);
    k_sqrt_isqrt<<<chB, TPB, 0, stream>>>(meanP, mean_sq, mean_isq, Cn);
    // 6) batch_var and scale s = std / sqrt(var + eps)
    hipMemsetAsync(var, 0, (size_t)Cn * sizeof(float), stream);
    k_var<<<(unsigned)((nmat + 7) / 8), 256, 0, stream>>>(XT, GT, var, nmat, Bn, Cn);
    k_scale<<<1, 32, 0, stream>>>(var, stdP, svec, Cn);
    // 7) Xn = mean_sq * (rm_isq X rm_isq)^s * mean_sq
    k_final<<<smB, TPB, 0, stream>>>(X, rm_isq, mean_sq, svec, (float*)d_out,
                                     nmat, Cn);
}